// GATWithAtt_15341623181498
// MI455X (gfx1250) — compile-verified
//
#include <hip/hip_runtime.h>
#include <hip/hip_bf16.h>

typedef __attribute__((ext_vector_type(2))) float v2f;
typedef __attribute__((ext_vector_type(8))) float v8f;
typedef __attribute__((ext_vector_type(4))) unsigned v4u;
typedef __attribute__((ext_vector_type(8))) unsigned v8u;

#define NEG_SLOPE 0.2f
#define BN_EPS 1e-5f

// ---------------- helpers ----------------

__device__ __forceinline__ unsigned encf(float f) {
    unsigned u = __float_as_uint(f);
    return (u & 0x80000000u) ? ~u : (u | 0x80000000u);
}
__device__ __forceinline__ float decf(unsigned u) {
    return (u & 0x80000000u) ? __uint_as_float(u & 0x7FFFFFFFu) : __uint_as_float(~u);
}
__device__ __forceinline__ void edge_sd(const long long* ei, long E, long e, int& s, int& d) {
    if (e < E) { s = (int)ei[e]; d = (int)ei[E + e]; }
    else       { s = d = (int)(e - E); }
}

// ---------------- init kernels ----------------

__global__ void zero_f32(float* p, long n) {
    long i = (long)blockIdx.x * blockDim.x + threadIdx.x;
    long st = (long)gridDim.x * blockDim.x;
    for (; i < n; i += st) p[i] = 0.0f;
}

__global__ void init_out_x2(float* out, const float* bias2, int n2) {
    int i = blockIdx.x * blockDim.x + threadIdx.x;
    if (i < n2) out[i] = bias2[i & 1];
}

// ---------------- layer 1: h1 = x @ W1 via fp32 WMMA + TDM-staged A tile ----
// x: [N,128], W1: [128,256], h1: [N,256].
// One workgroup (8 waves) computes a 16x256 output strip. The 16x128 x-tile is
// DMA'd into LDS once by the Tensor Data Mover (with 2-DWORD row padding to
// de-conflict LDS banks); each wave runs two interleaved 16x16x4 WMMA chains
// (col tiles wid and wid+8) sharing the same A fragment.
__global__ void gemm1_wmma(const float* __restrict__ x, const float* __restrict__ W1,
                           float* __restrict__ h1, int N) {
    __shared__ float axt[16 * 130];            // 16 rows x (128 + 2 pad) floats
    const int lane = threadIdx.x & 31;
    const int wid  = threadIdx.x >> 5;
    const int rt   = blockIdx.x;

    // ---- TDM: stage x tile [rows x 128] f32 into LDS (wave 0 issues) ----
    if (threadIdx.x < 32) {
        unsigned lds_off = (unsigned)(size_t)(void*)&axt[0];     // low 32b = LDS offset
        unsigned long long ga =
            (unsigned long long)(size_t)(x + (size_t)rt * 16 * 128);
        int rows = N - rt * 16; if (rows > 16) rows = 16;        // OOB rows read as 0

        // D# group0: count=1 | lds_addr | global_addr[56:0] | type=2
        v4u g0 = { 1u, lds_off, (unsigned)ga,
                   (unsigned)((ga >> 32) & 0x01FFFFFFu) | 0x80000000u };
        // D# group1: data_size=4B, pad_enable, pad_interval=128dw, pad_amount=2dw,
        //            tensor_dim0=128, tensor_dim1=rows, tile 128x16, stride0=128
        v8u g1 = { 0x20000u | (1u << 20) | (6u << 22) | (1u << 25),
                   128u << 16,                 // [47:32]=abar(0), [63:48]=tdim0.lo
                   ((unsigned)rows) << 16,     // [79:64]=tdim0.hi(0), [95:80]=tdim1.lo
                   128u << 16,                 // [111:96]=tdim1.hi(0), [127:112]=tile_dim0
                   16u,                        // [143:128]=tile_dim1, [159:144]=tile_dim2(0)
                   128u,                       // tensor_dim0_stride[31:0]
                   0u, 0u };
        asm volatile("tensor_load_to_lds %0, %1" :: "s"(g0), "s"(g1) : "memory");
        __builtin_amdgcn_s_wait_tensorcnt(0);
    }
    __syncthreads();

    const int m    = lane & 15;
    const int koff = (lane >> 4) << 1;          // lanes 0-15 -> K+0/1, 16-31 -> K+2/3
    const int col0 = wid * 16 + m;              // col tile wid
    const int col1 = (wid + 8) * 16 + m;        // col tile wid+8
    const float* ar = &axt[m * 130];

    v8f acc0 = {}, acc1 = {};
#pragma unroll
    for (int k0 = 0; k0 < 128; k0 += 4) {
        const int k = k0 + koff;
        v2f a, b0, b1;
        a.x  = ar[k];
        a.y  = ar[k + 1];
        b0.x = W1[(long)k * 256 + col0];
        b0.y = W1[(long)(k + 1) * 256 + col0];
        b1.x = W1[(long)k * 256 + col1];
        b1.y = W1[(long)(k + 1) * 256 + col1];
        acc0 = __builtin_amdgcn_wmma_f32_16x16x4_f32(
            false, a, false, b0, (short)0, acc0, false, false);
        acc1 = __builtin_amdgcn_wmma_f32_16x16x4_f32(
            false, a, false, b1, (short)0, acc1, false, false);
    }

    // D layout: VGPR v -> row = rt*16 + v + (lane>=16 ? 8 : 0), col = ct*16 + (lane&15)
    const int rbase = rt * 16 + ((lane >> 4) << 3);
#pragma unroll
    for (int v = 0; v < 8; ++v) {
        const int r = rbase + v;
        if (r < N) {
            h1[(long)r * 256 + col0] = acc0[v];
            h1[(long)r * 256 + col1] = acc1[v];
        }
    }
}

// ---------------- attention scalar features (layer 1) ----------------
__global__ void attn_feats1(const float* __restrict__ h1, const float* __restrict__ a_src,
                            const float* __restrict__ a_dst, float* es, float* ed, int N) {
    int idx = blockIdx.x * blockDim.x + threadIdx.x;
    if (idx >= N * 4) return;
    const int n = idx >> 2, hh = idx & 3;
    const float* hr = h1 + (long)n * 256 + hh * 64;
    const float* as = a_src + hh * 64;
    const float* ad = a_dst + hh * 64;
    float s = 0.f, d = 0.f;
#pragma unroll 8
    for (int i = 0; i < 64; ++i) { float v = hr[i]; s += v * as[i]; d += v * ad[i]; }
    es[idx] = s; ed[idx] = d;
}

// ---------------- edge pass 1: leaky-relu + segment max ----------------
__global__ void edge_max1(const long long* ei, long E, long ET,
                          const float* __restrict__ es, const float* __restrict__ ed,
                          float* e1, unsigned* emax) {
    long e = (long)blockIdx.x * blockDim.x + threadIdx.x;
    if (e >= ET) return;
    int s, d; edge_sd(ei, E, e, s, d);
#pragma unroll
    for (int h = 0; h < 4; ++h) {
        float v = es[s * 4 + h] + ed[d * 4 + h];
        v = v > 0.f ? v : NEG_SLOPE * v;
        e1[e * 4 + h] = v;
        atomicMax(&emax[d * 4 + h], encf(v));
    }
}

// ---------------- edge pass 2: exp + segment sum ----------------
__global__ void edge_exp1(const long long* ei, long E, long ET,
                          float* e1, const unsigned* emax, float* den) {
    long e = (long)blockIdx.x * blockDim.x + threadIdx.x;
    if (e >= ET) return;
    int s, d; edge_sd(ei, E, e, s, d);
#pragma unroll
    for (int h = 0; h < 4; ++h) {
        float mx = decf(emax[d * 4 + h]);
        float x = __expf(e1[e * 4 + h] - mx);
        e1[e * 4 + h] = x;
        atomicAdd(&den[d * 4 + h], x);
    }
}

// ---------------- weighted scatter aggregate: one wave per edge ----------------
__global__ void aggregate1(const long long* ei, long E, long ET,
                           const float* __restrict__ h1, const float* __restrict__ ex,
                           const float* __restrict__ den, float* agg) {
    const int lane = threadIdx.x & 31;
    long e = (long)blockIdx.x * 8 + (threadIdx.x >> 5);
    if (e >= ET) return;                     // uniform per wave
    int s, d; edge_sd(ei, E, e, s, d);
    float al[4];
#pragma unroll
    for (int h = 0; h < 4; ++h) al[h] = ex[e * 4 + h] / (den[d * 4 + h] + 1e-16f);
    const float* hs = h1 + (long)s * 256;
    float* od = agg + (long)d * 256;
#pragma unroll
    for (int i = 0; i < 8; ++i) {
        const int c = lane + i * 32;          // coalesced 128B per wave
        atomicAdd(&od[c], hs[c] * al[c >> 6]);
    }
}

// ---------------- BatchNorm statistics ----------------
__global__ void bn_stats(const float* __restrict__ agg, const float* __restrict__ bias1,
                         float* sum, float* sumsq, int N) {
    const int c = threadIdx.x;               // 256 channels
    const float b = bias1[c];
    float s = 0.f, s2 = 0.f;
    for (int r = blockIdx.x; r < N; r += gridDim.x) {
        float y = agg[(long)r * 256 + c] + b;
        s += y; s2 += y * y;
    }
    atomicAdd(&sum[c], s);
    atomicAdd(&sumsq[c], s2);
}

// ---------------- BN apply + ELU -> x1 ----------------
__global__ void bn_apply_elu(const float* __restrict__ agg, const float* __restrict__ bias1,
                             const float* __restrict__ gamma, const float* __restrict__ beta,
                             const float* __restrict__ sum, const float* __restrict__ sumsq,
                             float* x1, int N) {
    long idx = (long)blockIdx.x * blockDim.x + threadIdx.x;
    if (idx >= (long)N * 256) return;
    const int c = (int)(idx & 255);
    const float invN = 1.0f / (float)N;
    const float mu = sum[c] * invN;
    const float var = sumsq[c] * invN - mu * mu;
    const float rs = rsqrtf(var + BN_EPS);
    float y = (agg[idx] + bias1[c] - mu) * rs * gamma[c] + beta[c];
    x1[idx] = y > 0.f ? y : (__expf(y) - 1.0f);   // ELU
}

// ---------------- layer 2 node features: wave-per-node GEMV (256x2) ----------------
__global__ void l2_feats(const float* __restrict__ x1, const float* __restrict__ W2,
                         const float* __restrict__ as2, const float* __restrict__ ad2,
                         float* h2, float* es2, float* ed2, int N) {
    const int lane = threadIdx.x & 31;
    const int n = blockIdx.x * 8 + (threadIdx.x >> 5);
    if (n >= N) return;
    const float* xr = x1 + (long)n * 256;
    float a0 = 0.f, a1 = 0.f;
#pragma unroll
    for (int i = lane; i < 256; i += 32) {
        float v = xr[i];
        a0 += v * W2[i * 2];
        a1 += v * W2[i * 2 + 1];
    }
#pragma unroll
    for (int m = 16; m > 0; m >>= 1) {
        a0 += __shfl_xor(a0, m, 32);
        a1 += __shfl_xor(a1, m, 32);
    }
    if (lane == 0) {
        h2[n * 2] = a0; h2[n * 2 + 1] = a1;
        es2[n] = a0 * as2[0] + a1 * as2[1];
        ed2[n] = a0 * ad2[0] + a1 * ad2[1];
    }
}

// ---------------- layer 2 edge passes ----------------
__global__ void edge_max2(const long long* ei, long E, long ET,
                          const float* es2, const float* ed2, float* e2, unsigned* emax2) {
    long e = (long)blockIdx.x * blockDim.x + threadIdx.x;
    if (e >= ET) return;
    int s, d; edge_sd(ei, E, e, s, d);
    float v = es2[s] + ed2[d];
    v = v > 0.f ? v : NEG_SLOPE * v;
    e2[e] = v;
    atomicMax(&emax2[d], encf(v));
}

__global__ void edge_exp2(const long long* ei, long E, long ET,
                          float* e2, const unsigned* emax2, float* den2) {
    long e = (long)blockIdx.x * blockDim.x + threadIdx.x;
    if (e >= ET) return;
    int s, d; edge_sd(ei, E, e, s, d);
    float x = __expf(e2[e] - decf(emax2[d]));
    e2[e] = x;
    atomicAdd(&den2[d], x);
}

__global__ void edge_final2(const long long* ei, long E, long ET,
                            const float* __restrict__ h2, const float* __restrict__ ex2,
                            const float* __restrict__ den2, float* out_x2, float* out_att) {
    long e = (long)blockIdx.x * blockDim.x + threadIdx.x;
    if (e >= ET) return;
    int s, d; edge_sd(ei, E, e, s, d);
    float al = ex2[e] / (den2[d] + 1e-16f);
    out_att[e] = al;
    atomicAdd(&out_x2[d * 2],     h2[s * 2]     * al);
    atomicAdd(&out_x2[d * 2 + 1], h2[s * 2 + 1] * al);
}

// ---------------- launch ----------------

extern "C" void kernel_launch(void* const* d_in, const int* in_sizes, int n_in,
                              void* d_out, int out_size, void* d_ws, size_t ws_size,
                              hipStream_t stream) {
    const float*     x    = (const float*)d_in[0];
    const long long* ei   = (const long long*)d_in[1];
    const float*     W1   = (const float*)d_in[2];
    const float*     as1  = (const float*)d_in[3];
    const float*     ad1  = (const float*)d_in[4];
    const float*     b1   = (const float*)d_in[5];
    const float*     gam  = (const float*)d_in[6];
    const float*     bet  = (const float*)d_in[7];
    const float*     W2   = (const float*)d_in[8];
    const float*     as2  = (const float*)d_in[9];
    const float*     ad2  = (const float*)d_in[10];
    const float*     b2   = (const float*)d_in[11];

    const int  N  = in_sizes[0] / 128;     // 50000
    const long E  = in_sizes[1] / 2;       // 800000
    const long ET = E + N;                 // edges + self loops

    // workspace layout (floats)
    float*    h1    = (float*)d_ws;                  // N*256 (reused as x1)
    float*    agg   = h1  + (size_t)N * 256;         // N*256
    float*    es1   = agg + (size_t)N * 256;         // N*4
    float*    ed1   = es1 + (size_t)N * 4;           // N*4
    unsigned* emax1 = (unsigned*)(ed1 + (size_t)N * 4);   // N*4
    float*    den1  = (float*)(emax1 + (size_t)N * 4);    // N*4 (contiguous with emax1)
    float*    e1    = den1 + (size_t)N * 4;          // ET*4
    float*    bsum  = e1   + (size_t)ET * 4;         // 256
    float*    bsq   = bsum + 256;                    // 256
    float*    h2    = bsq  + 256;                    // N*2
    float*    es2   = h2   + (size_t)N * 2;          // N
    float*    ed2   = es2  + (size_t)N;              // N
    unsigned* emax2 = (unsigned*)(ed2 + (size_t)N);  // N
    float*    den2  = (float*)(emax2 + (size_t)N);   // N (contiguous with emax2)
    float*    e2    = den2 + (size_t)N;              // ET

    float* out_x2  = (float*)d_out;            // [N,2]
    float* out_att = out_x2 + (size_t)N * 2;   // [ET,1]

    // ---- init accumulators (every call: replay-safe) ----
    zero_f32<<<2048, 256, 0, stream>>>(agg, (long)N * 256);
    zero_f32<<<256, 256, 0, stream>>>((float*)emax1, (long)N * 8);   // emax1 + den1
    zero_f32<<<2, 256, 0, stream>>>(bsum, 512);                      // bsum + bsq
    zero_f32<<<256, 256, 0, stream>>>((float*)emax2, (long)N * 2);   // emax2 + den2
    init_out_x2<<<(2 * N + 255) / 256, 256, 0, stream>>>(out_x2, b2, 2 * N);

    // ---- layer 1 ----
    gemm1_wmma<<<(N + 15) / 16, 256, 0, stream>>>(x, W1, h1, N);
    attn_feats1<<<(N * 4 + 255) / 256, 256, 0, stream>>>(h1, as1, ad1, es1, ed1, N);

    const int egrid = (int)((ET + 255) / 256);
    edge_max1<<<egrid, 256, 0, stream>>>(ei, E, ET, es1, ed1, e1, emax1);
    edge_exp1<<<egrid, 256, 0, stream>>>(ei, E, ET, e1, emax1, den1);
    aggregate1<<<(int)((ET + 7) / 8), 256, 0, stream>>>(ei, E, ET, h1, e1, den1, agg);

    // ---- BN + ELU ----
    bn_stats<<<512, 256, 0, stream>>>(agg, b1, bsum, bsq, N);
    bn_apply_elu<<<(int)(((long)N * 256 + 255) / 256), 256, 0, stream>>>(
        agg, b1, gam, bet, bsum, bsq, /*x1=*/h1, N);

    // ---- layer 2 ----
    l2_feats<<<(N + 7) / 8, 256, 0, stream>>>(h1, W2, as2, ad2, h2, es2, ed2, N);
    edge_max2<<<egrid, 256, 0, stream>>>(ei, E, ET, es2, ed2, e2, emax2);
    edge_exp2<<<egrid, 256, 0, stream>>>(ei, E, ET, e2, emax2, den2);
    edge_final2<<<egrid, 256, 0, stream>>>(ei, E, ET, h2, e2, den2, out_x2, out_att);
}